// DREAMCell_27865747817147
// MI455X (gfx1250) — compile-verified
//
#include <hip/hip_runtime.h>
#include <math.h>

// ---- problem constants (B,I,H,R) = (64,512,2048,64) ----
#define B_  64
#define I_  512
#define H_  2048
#define R_  64

#define TAU0_    0.5f
#define ALPHA_   0.1f
#define GAMMA_   0.1f
#define LAMBDA_  0.01f
#define AFS_     1.0f
#define TARGET_  1.0f
#define DT_      0.1f
#define EMA_     0.05f
#define HAB_     0.001f
#define MIN_TAU_ 0.01f
#define MAX_TAU_ 50.0f
#define EPS_     1e-6f
#define TWO_PI_E_ 17.079468445347132f   // 2*pi*e

typedef __attribute__((ext_vector_type(2))) float v2f;
typedef __attribute__((ext_vector_type(8))) float v8f;

// D = A(16x4 f32) * B(4x16 f32) + C(16x16 f32), exact fp32 — matches reference math.
__device__ __forceinline__ v8f wmma_k4(v2f a, v2f b, v8f c) {
    return __builtin_amdgcn_wmma_f32_16x16x4_f32(false, a, false, b,
                                                 (short)0, c, false, false);
}

// ---------------- K1: per-row ||x||, mean(error_var), zero accumulators ----
__global__ void k_rowstats(const float* __restrict__ x, const float* __restrict__ evar,
                           float* __restrict__ xmag, float* __restrict__ varmean,
                           float* __restrict__ errnorm2, float* __restrict__ usumsq) {
    const int b = blockIdx.x;
    const int t = threadIdx.x;   // 256
    float sx = 0.f, sv = 0.f;
    for (int i = t; i < I_; i += 256) {
        float xv = x[b * I_ + i];
        sx += xv * xv;
        sv += evar[b * I_ + i];
    }
    __shared__ float smx[256], smv[256];
    smx[t] = sx; smv[t] = sv;
    __syncthreads();
    for (int s = 128; s > 0; s >>= 1) {
        if (t < s) { smx[t] += smx[t + s]; smv[t] += smv[t + s]; }
        __syncthreads();
    }
    if (t == 0) {
        xmag[b]     = sqrtf(smx[0]);
        varmean[b]  = smv[0] / (float)I_;
        errnorm2[b] = 0.f;
        usumsq[b]   = 0.f;
    }
}

// ---------------- K2: error = x - tanh(h@C^T)*||x||  (WMMA, M=64,N=512,K=2048)
// Each wave: 16x64 strip. Pipelined; first step peeled so the steady loop has
// an EVEN trip count (510) -> unroll-2 ping-pongs prefetch regs without movs.
__global__ void k_gemm_err(const float* __restrict__ h, const float* __restrict__ C,
                           const float* __restrict__ x, const float* __restrict__ emean,
                           const float* __restrict__ evar, const float* __restrict__ xmag,
                           float* __restrict__ err, float* __restrict__ xnorm,
                           float* __restrict__ errnorm2,
                           float* __restrict__ out_em, float* __restrict__ out_ev) {
    const int lane = threadIdx.x & 31;
    const int wid  = (blockIdx.x * blockDim.x + threadIdx.x) >> 5;  // 0..31
    const int mt = wid & 3;    // B tile
    const int ng = wid >> 2;   // group of 64 cols in I (0..7)
    const int lh = lane & 15;
    const int kh = lane >> 4;  // which K half

    v8f acc[4] = {v8f{}, v8f{}, v8f{}, v8f{}};
    const float* Arow = h + (size_t)(mt * 16 + lh) * H_ + kh * 2;
    const float* Brow[4];
#pragma unroll
    for (int j = 0; j < 4; ++j)
        Brow[j] = C + (size_t)(ng * 64 + j * 16 + lh) * H_ + kh * 2;  // B[k][n]=C[n*H+k]

    // peel step 0 (and issue step-1 prefetch) so the steady loop trip count is even
    v2f a_c = *(const v2f*)(Arow + 4);
    v2f b_c[4];
    {
        v2f a0 = *(const v2f*)(Arow);
        v2f b0[4];
#pragma unroll
        for (int j = 0; j < 4; ++j) b0[j] = *(const v2f*)(Brow[j]);
#pragma unroll
        for (int j = 0; j < 4; ++j) b_c[j] = *(const v2f*)(Brow[j] + 4);
#pragma unroll
        for (int j = 0; j < 4; ++j) acc[j] = wmma_k4(a0, b0[j], acc[j]);
    }

#pragma unroll 2
    for (int k0 = 8; k0 < H_; k0 += 4) {   // 510 iterations (even)
        v2f a_n = *(const v2f*)(Arow + k0);
        v2f b_n[4];
#pragma unroll
        for (int j = 0; j < 4; ++j) b_n[j] = *(const v2f*)(Brow[j] + k0);
#pragma unroll
        for (int j = 0; j < 4; ++j) acc[j] = wmma_k4(a_c, b_c[j], acc[j]);
        a_c = a_n;
#pragma unroll
        for (int j = 0; j < 4; ++j) b_c[j] = b_n[j];
    }
#pragma unroll
    for (int j = 0; j < 4; ++j) acc[j] = wmma_k4(a_c, b_c[j], acc[j]);   // drain

#pragma unroll
    for (int j = 0; j < 4; ++j) {
        const int n = ng * 64 + j * 16 + lh;
#pragma unroll
        for (int i = 0; i < 8; ++i) {
            const int b = mt * 16 + i + kh * 8;
            const float xm   = xmag[b];
            const float pred = tanhf(acc[j][i]) * xm;
            const float xv   = x[b * I_ + n];
            const float e    = xv - pred;
            err[b * I_ + n]  = e;
            float xn = xv / (xm + EPS_);
            xn = fminf(fmaxf(xn, -1.f), 1.f);
            xnorm[b * I_ + n] = xn;
            const float em = (1.f - EMA_) * emean[b * I_ + n] + EMA_ * e;
            out_em[b * I_ + n] = em;
            const float d = e - em;
            out_ev[b * I_ + n] = (1.f - EMA_) * evar[b * I_ + n] + EMA_ * d * d;
            // row-wise ||error||^2 : reduce across the 16 lanes holding this row's cols
            float v = e * e;
#pragma unroll
            for (int off = 8; off >= 1; off >>= 1) v += __shfl_down(v, off, 16);
            if (lh == 0) atomicAdd(&errnorm2[b], v);
        }
    }
}

// ---------------- K3: per-batch scalars ----------------
__global__ void k_scalars(const float* __restrict__ xmag, const float* __restrict__ varmean,
                          const float* __restrict__ errnorm2, const float* __restrict__ atau_in,
                          const float* __restrict__ eta_p, const float* __restrict__ tausys_p,
                          const float* __restrict__ ltc_p,
                          float* __restrict__ out_atau, float* __restrict__ out_s,
                          float* __restrict__ coefH, float* __restrict__ coefT,
                          float* __restrict__ aU, float* __restrict__ cU) {
    const int b = threadIdx.x;
    if (b >= B_) return;
    const float rel = sqrtf(errnorm2[b]) / (xmag[b] + EPS_);
    float ent = 0.5f * logf(TWO_PI_E_ * (varmean[b] + EPS_));
    ent = fminf(fmaxf(ent, 0.f), 2.f);
    const float natau = fminf((1.f - HAB_) * atau_in[b] + HAB_ * rel, 0.8f);
    out_atau[b] = natau;
    const float cls = TAU0_ * (1.f + ALPHA_ * ent);
    const float eff = 0.3f * cls + 0.7f * natau;
    const float s = 1.f / (1.f + expf(-(rel - eff) / GAMMA_));
    out_s[b] = s;
    const float tausys = tausys_p[0];
    const float tc = fmaxf(tausys, MIN_TAU_);
    const float td = tc / (1.f + s * expf(ltc_p[0]));
    const float te = fminf(fmaxf(td, MIN_TAU_), MAX_TAU_);
    const float dt = fminf(fmaxf(DT_ / (te + DT_), 0.01f), 0.5f);
    const float ul = 1.f / (1.f + expf(-(tausys - 0.01f) * 100.f));
    coefH[b] = ul * (1.f - dt) + (1.f - ul) * 0.05f;   // weight on h
    coefT[b] = ul * dt + (1.f - ul) * 0.95f;           // weight on h_target
    const float lam = LAMBDA_ * (1.f + AFS_ * s);
    aU[b] = 1.f - lam * DT_;
    cU[b] = DT_ * eta_p[0] * s;
}

// ---------------- K4: eV = error @ V  (WMMA, M=64,N=64,K=512; one block, 16 waves)
__global__ void k_gemm_ev(const float* __restrict__ err, const float* __restrict__ V,
                          float* __restrict__ eV) {
    const int lane = threadIdx.x & 31;
    const int wid  = threadIdx.x >> 5;   // 0..15
    const int mt = wid & 3, nt = wid >> 2;
    const int lh = lane & 15, kh = lane >> 4;
    v8f acc = {};
    const float* A = err + (size_t)(mt * 16 + lh) * I_ + kh * 2;
    const int n = nt * 16 + lh;
    const float* Bp = V + (size_t)(kh * 2) * R_ + n;

    // peel step 0 so steady loop has even trips (126)
    v2f a_c = *(const v2f*)(A + 4);
    v2f b_c;
    {
        v2f a0 = *(const v2f*)(A);
        v2f b0; b0.x = Bp[0]; b0.y = Bp[R_];
        b_c.x = Bp[4 * R_]; b_c.y = Bp[5 * R_];
        acc = wmma_k4(a0, b0, acc);
    }
#pragma unroll 2
    for (int k0 = 8; k0 < I_; k0 += 4) {   // 126 iterations (even)
        v2f a_n = *(const v2f*)(A + k0);
        v2f b_n;
        b_n.x = Bp[(size_t)k0 * R_];
        b_n.y = Bp[(size_t)(k0 + 1) * R_];
        acc = wmma_k4(a_c, b_c, acc);
        a_c = a_n; b_c = b_n;
    }
    acc = wmma_k4(a_c, b_c, acc);
#pragma unroll
    for (int i = 0; i < 8; ++i) {
        const int b = mt * 16 + i + kh * 8;
        eV[b * R_ + n] = acc[i];
    }
}

// ---------------- K5: h_new (dual WMMA GEMM, M=64,N=2048,K=512) -------------
// Each wave: 16x64 strip of both GEMMs (8 accumulators), pipelined, peeled.
__global__ void k_gemm_h(const float* __restrict__ xnorm, const float* __restrict__ err,
                         const float* __restrict__ Bm, const float* __restrict__ W,
                         const float* __restrict__ h, const float* __restrict__ surprise,
                         const float* __restrict__ coefH, const float* __restrict__ coefT,
                         float* __restrict__ out_h) {
    const int lane = threadIdx.x & 31;
    const int wid  = (blockIdx.x * blockDim.x + threadIdx.x) >> 5;  // 0..127
    const int mt = wid & 3;        // B tile
    const int ng = wid >> 2;       // group of 64 cols in H (0..31)
    const int lh = lane & 15, kh = lane >> 4;

    v8f accB[4] = {v8f{}, v8f{}, v8f{}, v8f{}};
    v8f accE[4] = {v8f{}, v8f{}, v8f{}, v8f{}};
    const float* A1 = xnorm + (size_t)(mt * 16 + lh) * I_ + kh * 2;
    const float* A2 = err   + (size_t)(mt * 16 + lh) * I_ + kh * 2;
    const float* B1[4];
    const float* B2[4];
#pragma unroll
    for (int j = 0; j < 4; ++j) {
        B1[j] = Bm + (size_t)(ng * 64 + j * 16 + lh) * I_ + kh * 2;  // B[k][n]=Bm[n*I+k]
        B2[j] = W  + (size_t)(ng * 64 + j * 16 + lh) * I_ + kh * 2;
    }

    // peel step 0; prefetch step 1
    v2f a1c = *(const v2f*)(A1 + 4), a2c = *(const v2f*)(A2 + 4);
    v2f b1c[4], b2c[4];
    {
        v2f a10 = *(const v2f*)(A1), a20 = *(const v2f*)(A2);
        v2f b10[4], b20[4];
#pragma unroll
        for (int j = 0; j < 4; ++j) {
            b10[j] = *(const v2f*)(B1[j]);
            b20[j] = *(const v2f*)(B2[j]);
            b1c[j] = *(const v2f*)(B1[j] + 4);
            b2c[j] = *(const v2f*)(B2[j] + 4);
        }
#pragma unroll
        for (int j = 0; j < 4; ++j) {
            accB[j] = wmma_k4(a10, b10[j], accB[j]);
            accE[j] = wmma_k4(a20, b20[j], accE[j]);
        }
    }

#pragma unroll 2
    for (int k0 = 8; k0 < I_; k0 += 4) {   // 126 iterations (even)
        v2f a1n = *(const v2f*)(A1 + k0);
        v2f a2n = *(const v2f*)(A2 + k0);
        v2f b1n[4], b2n[4];
#pragma unroll
        for (int j = 0; j < 4; ++j) {
            b1n[j] = *(const v2f*)(B1[j] + k0);
            b2n[j] = *(const v2f*)(B2[j] + k0);
        }
#pragma unroll
        for (int j = 0; j < 4; ++j) {
            accB[j] = wmma_k4(a1c, b1c[j], accB[j]);
            accE[j] = wmma_k4(a2c, b2c[j], accE[j]);
        }
        a1c = a1n; a2c = a2n;
#pragma unroll
        for (int j = 0; j < 4; ++j) { b1c[j] = b1n[j]; b2c[j] = b2n[j]; }
    }
#pragma unroll
    for (int j = 0; j < 4; ++j) {                                    // drain
        accB[j] = wmma_k4(a1c, b1c[j], accB[j]);
        accE[j] = wmma_k4(a2c, b2c[j], accE[j]);
    }

#pragma unroll
    for (int j = 0; j < 4; ++j) {
        const int n = ng * 64 + j * 16 + lh;
#pragma unroll
        for (int i = 0; i < 8; ++i) {
            const int b = mt * 16 + i + kh * 8;
            const float hv = h[(size_t)b * H_ + n];
            const float ie = 0.7f * hv + 0.2f * accB[j][i] + 0.3f * surprise[b] * accE[j][i];
            const float ht = tanhf(ie);
            out_h[(size_t)b * H_ + n] = coefH[b] * hv + coefT[b] * ht;
        }
    }
}

// ---------------- K6: U_pre = a*U + (1-a)*U_target + c*h⊗eV; per-b sum(U^2) --
__global__ void k_uupdate(const float* __restrict__ U, const float* __restrict__ Ut,
                          const float* __restrict__ h, const float* __restrict__ eV,
                          const float* __restrict__ aU, const float* __restrict__ cU,
                          float* __restrict__ outU, float* __restrict__ usumsq) {
    const int b = blockIdx.y;
    const int idx = blockIdx.x * blockDim.x + threadIdx.x;  // 0 .. H*R-1
    const int hh = idx >> 6;      // / R_
    const int r  = idx & (R_ - 1);
    const size_t g = (size_t)b * H_ * R_ + idx;
    const float a = aU[b], c = cU[b];
    const float val = a * U[g] + (1.f - a) * Ut[g]
                    + c * h[(size_t)b * H_ + hh] * eV[b * R_ + r];
    outU[g] = val;
    __shared__ float sm[256];
    sm[threadIdx.x] = val * val;
    __syncthreads();
    for (int s = 128; s > 0; s >>= 1) {
        if (threadIdx.x < s) sm[threadIdx.x] += sm[threadIdx.x + s];
        __syncthreads();
    }
    if (threadIdx.x == 0) atomicAdd(&usumsq[b], sm[0]);
}

// ---------------- K7: Frobenius rescale ----------------
__global__ void k_uscale(float* __restrict__ outU, const float* __restrict__ usumsq) {
    const int b = blockIdx.y;
    const int idx = blockIdx.x * blockDim.x + threadIdx.x;
    const float scale = fminf(TARGET_ / (sqrtf(usumsq[b]) + EPS_), 1.5f);
    const size_t g = (size_t)b * H_ * R_ + idx;
    outU[g] *= scale;
}

extern "C" void kernel_launch(void* const* d_in, const int* in_sizes, int n_in,
                              void* d_out, int out_size, void* d_ws, size_t ws_size,
                              hipStream_t stream) {
    (void)in_sizes; (void)n_in; (void)out_size; (void)ws_size;
    const float* x      = (const float*)d_in[0];
    const float* h      = (const float*)d_in[1];
    const float* U      = (const float*)d_in[2];
    const float* Ut     = (const float*)d_in[3];
    const float* emean  = (const float*)d_in[4];
    const float* evar   = (const float*)d_in[5];
    const float* atau   = (const float*)d_in[6];
    const float* C      = (const float*)d_in[7];
    const float* W      = (const float*)d_in[8];
    const float* Bm     = (const float*)d_in[9];
    const float* V      = (const float*)d_in[10];
    const float* eta    = (const float*)d_in[11];
    const float* tausys = (const float*)d_in[12];
    const float* ltc    = (const float*)d_in[13];

    // output layout (flat, return order)
    float* out      = (float*)d_out;
    float* out_h    = out;                          // 64*2048
    float* out_U    = out + 131072;                 // 64*2048*64
    float* out_em   = out + 131072 + 8388608;       // 64*512
    float* out_ev   = out_em + 32768;               // 64*512
    float* out_atau = out_ev + 32768;               // 64
    float* out_s    = out_atau + 64;                // 64

    // workspace layout (floats)
    float* ws        = (float*)d_ws;
    float* err       = ws;               // 32768
    float* xnorm     = ws + 32768;       // 32768
    float* xmag      = ws + 65536;       // 64
    float* varmean   = ws + 65600;       // 64
    float* errnorm2  = ws + 65664;       // 64
    float* eV        = ws + 65728;       // 4096
    float* coefH     = ws + 69824;       // 64
    float* coefT     = coefH + 64;       // 64
    float* aU        = coefT + 64;       // 64
    float* cU        = aU + 64;          // 64
    float* usq       = cU + 64;          // 64

    k_rowstats<<<B_, 256, 0, stream>>>(x, evar, xmag, varmean, errnorm2, usq);
    // 32 waves (4 M-tiles x 8 col-groups), 4 waves/block
    k_gemm_err<<<8, 128, 0, stream>>>(h, C, x, emean, evar, xmag,
                                      err, xnorm, errnorm2, out_em, out_ev);
    k_scalars<<<1, 64, 0, stream>>>(xmag, varmean, errnorm2, atau, eta, tausys, ltc,
                                    out_atau, out_s, coefH, coefT, aU, cU);
    k_gemm_ev<<<1, 512, 0, stream>>>(err, V, eV);
    // 128 waves (4 M-tiles x 32 col-groups), 8 waves/block
    k_gemm_h<<<16, 256, 0, stream>>>(xnorm, err, Bm, W, h, out_s, coefH, coefT, out_h);
    dim3 gU(H_ * R_ / 256, B_);
    k_uupdate<<<gU, 256, 0, stream>>>(U, Ut, h, eV, aU, cU, out_U, usq);
    k_uscale<<<gU, 256, 0, stream>>>(out_U, usq);
}